// Multi_Block_88098369176509
// MI455X (gfx1250) — compile-verified
//
#include <hip/hip_runtime.h>

// CDNA5 / gfx1250, wave32. Matrix math via V_WMMA_F32_16X16X4_F32.
typedef float v2f __attribute__((ext_vector_type(2)));
typedef float v4f __attribute__((ext_vector_type(4)));
typedef float v8f __attribute__((ext_vector_type(8)));

#define C_IN   16
#define C_OUT  16
#define HDIM   96
#define WDIM   96
#define PLANE  (HDIM * WDIM)       // 9216
#define NP     9                   // 3x3 taps
#define BATCH  4
#define TILES_PER_ROW   (WDIM / 16)            // 6
#define TILES_PER_BATCH (HDIM * TILES_PER_ROW) // 576
#define TOTAL_TILES     (BATCH * TILES_PER_BATCH) // 2304
#define WAVES_PER_BLOCK 8
#define KMAT_FLOATS (BATCH * NP * C_IN * C_OUT)   // 9216
#define WT_FLOATS   (NP * C_OUT * NP * C_IN)      // 144*144 = 20736

__device__ __forceinline__ v8f wmma_f32_k4(v2f a, v2f b, v8f c) {
    // (neg_a, A, neg_b, B, c_mod, C, reuse_a, reuse_b)
    return __builtin_amdgcn_wmma_f32_16x16x4_f32(
        false, a, false, b, (short)0, c, false, false);
}

__device__ __forceinline__ int iclamp(int v, int lim) {
    int t = v < 0 ? 0 : v;
    return t > lim ? lim : t;
}

// ---------------------------------------------------------------------------
// Prep 1: Kt[b][p][o][c] = sum_ci cf[b,ci]*W_CF[p*256+c*16+o, ci] + b_CF
// (transposed so the WMMA B-fragment c-pair is a contiguous float2)
// ---------------------------------------------------------------------------
__global__ void __launch_bounds__(256)
prep_kmat(const float* __restrict__ cf, const float* __restrict__ W_CF,
          const float* __restrict__ b_CF, float* __restrict__ Kt) {
    int idx = blockIdx.x * 256 + threadIdx.x;          // 0 .. 9215
    int bb  = idx / (NP * C_IN * C_OUT);
    int e   = idx - bb * (NP * C_IN * C_OUT);          // p*256 + c*16 + o
    int p   = e >> 8;
    int r   = e & 255;
    int c   = r >> 4;
    int o   = r & 15;
    const float* cfb  = cf   + bb * C_IN;
    const float* wrow = W_CF + e * C_IN;
    float acc = b_CF[e];
#pragma unroll
    for (int ci = 0; ci < C_IN; ++ci) acc = fmaf(cfb[ci], wrow[ci], acc);
    Kt[((bb * NP + p) * C_OUT + o) * C_IN + c] = acc;
}

// ---------------------------------------------------------------------------
// Prep 2: Wt[p][t][o][ci] = W_M[oc=p*16+o, k=ci*9+t]
// (tap-major K-order, ci contiguous -> float2 B-fragment loads)
// ---------------------------------------------------------------------------
__global__ void __launch_bounds__(256)
prep_wt(const float* __restrict__ W_M, float* __restrict__ Wt) {
    int idx = blockIdx.x * 256 + threadIdx.x;          // 0 .. 20735 (81*256 exact)
    int oc  = idx / (NP * C_IN);                       // 0..143
    int k   = idx - oc * (NP * C_IN);                  // ci*9 + t
    int p   = oc >> 4;
    int o   = oc & 15;
    int ci  = k / NP;
    int t   = k - ci * NP;
    Wt[((p * NP + t) * C_OUT + o) * C_IN + ci] = W_M[idx];
}

// ---------------------------------------------------------------------------
// Fused main kernel: one wave per 16-pixel tile.
// Per p: 36 WMMAs build the conv tile (K tap-major), 4 WMMAs build the
// patch-GEMM tile, 8 packed FMAs accumulate their elementwise product.
// ---------------------------------------------------------------------------
__global__ void __launch_bounds__(32 * WAVES_PER_BLOCK)
fused_kernel(const float* __restrict__ feature, const float* __restrict__ filters,
             const float* __restrict__ b_M, const float* __restrict__ Kt,
             const float* __restrict__ Wt, float* __restrict__ out) {
    const int lane = threadIdx.x & 31;
    const int half = lane >> 4;     // K-pair selector: k = 2*half + j
    const int lo   = lane & 15;     // m (A rows) / n,o (B cols, C/D cols)

    const int tile = blockIdx.x * WAVES_PER_BLOCK + (threadIdx.x >> 5); // exact 2304
    const int bb = tile / TILES_PER_BATCH;
    const int t0 = tile - bb * TILES_PER_BATCH;
    const int h  = t0 / TILES_PER_ROW;
    const int w0 = (t0 - h * TILES_PER_ROW) * 16;

    // half*2*PLANE folded in: A element j lives at base + (cs + j)*PLANE
    const float* filt = filters + bb * C_IN * PLANE + half * 2 * PLANE;
    const float* feat = feature + bb * C_IN * PLANE + half * 2 * PLANE;
    // per-lane B row bases (pair offset 2*half folded in)
    const float* wt0 = Wt + lo * C_IN + half * 2;
    const float* kt0 = Kt + (bb * NP) * (C_IN * C_OUT) + lo * C_IN + half * 2;

    v8f acc = {0.f, 0.f, 0.f, 0.f, 0.f, 0.f, 0.f, 0.f};

#pragma unroll 1
    for (int p = 0; p < NP; ++p) {
        // ---- M tile: conv3x3 implicit GEMM, bias preloaded ----------------
        const float bias = b_M[p * C_OUT + lo];
        v8f cm = {bias, bias, bias, bias, bias, bias, bias, bias};
        const float* wtp = wt0 + p * (NP * C_IN * C_OUT);

#pragma unroll
        for (int t = 0; t < NP; ++t) {           // taps: ki,kj constants
            const int ki = t / 3, kj = t - (t / 3) * 3;
            const int hh = h + ki - 1;
            const int ww = w0 + lo + kj - 1;
            const bool ok = ((unsigned)hh < HDIM) & ((unsigned)ww < WDIM);
            const float* ab = filt + iclamp(hh, HDIM - 1) * WDIM
                                   + iclamp(ww, WDIM - 1);
            const float* wrow = wtp + t * (C_IN * C_OUT);
#pragma unroll
            for (int cs = 0; cs < C_IN; cs += 4) {
                v2f bf = *(const v2f*)(wrow + cs);
                float a0 = ab[(cs + 0) * PLANE];   // unconditional clamped load
                float a1 = ab[(cs + 1) * PLANE];
                v2f af;
                af[0] = ok ? a0 : 0.0f;            // v_cndmask, no exec dance
                af[1] = ok ? a1 : 0.0f;
                cm = wmma_f32_k4(af, bf, cm);
            }
        }

        // ---- T tile: (16 pixels x 16 c) @ Kmat[b,p] -----------------------
        const int di = p / 3 - 1;
        const int dj = p - (p / 3) * 3 - 1;
        const int hh2 = h + di;
        const int ww2 = w0 + lo + dj;
        const bool okA = ((unsigned)hh2 < HDIM) & ((unsigned)ww2 < WDIM);
        const float* fb = feat + iclamp(hh2, HDIM - 1) * WDIM
                               + iclamp(ww2, WDIM - 1);
        const float* krow = kt0 + p * (C_IN * C_OUT);

        v8f ct = {0.f, 0.f, 0.f, 0.f, 0.f, 0.f, 0.f, 0.f};
#pragma unroll
        for (int cs = 0; cs < C_IN; cs += 4) {
            v2f bf = *(const v2f*)(krow + cs);
            float a0 = fb[(cs + 0) * PLANE];
            float a1 = fb[(cs + 1) * PLANE];
            v2f af;
            af[0] = okA ? a0 : 0.0f;
            af[1] = okA ? a1 : 0.0f;
            ct = wmma_f32_k4(af, bf, ct);
        }

        // ---- elementwise product-accumulate (identical C/D layouts) -------
#pragma unroll
        for (int v = 0; v < 8; ++v) acc[v] = fmaf(cm[v], ct[v], acc[v]);
    }

    // D layout: lane -> o = lo, VGPR v -> pixel m = v + 8*half
    // => 8 consecutive w positions at one channel: two aligned float4 stores.
    float* op = out + ((bb * C_OUT + lo) * HDIM + h) * WDIM + w0 + half * 8;
    *(v4f*)(op + 0) = (v4f){acc[0], acc[1], acc[2], acc[3]};
    *(v4f*)(op + 4) = (v4f){acc[4], acc[5], acc[6], acc[7]};
}

extern "C" void kernel_launch(void* const* d_in, const int* in_sizes, int n_in,
                              void* d_out, int out_size, void* d_ws, size_t ws_size,
                              hipStream_t stream) {
    const float* feature = (const float*)d_in[0]; // (4,16,96,96)
    const float* filters = (const float*)d_in[1]; // (4,16,96,96)
    const float* cf      = (const float*)d_in[2]; // (4,16,1,1)
    const float* W_M     = (const float*)d_in[3]; // (144,16,3,3)
    const float* b_M     = (const float*)d_in[4]; // (144,)
    const float* W_CF    = (const float*)d_in[5]; // (2304,16,1,1)
    const float* b_CF    = (const float*)d_in[6]; // (2304,)
    float*       out     = (float*)d_out;         // (4,16,96,96)

    float* Kt = (float*)d_ws;                     // 9216 floats
    float* Wt = Kt + KMAT_FLOATS;                 // 20736 floats (total ~117 KB)

    prep_kmat<<<KMAT_FLOATS / 256, 256, 0, stream>>>(cf, W_CF, b_CF, Kt);
    prep_wt<<<WT_FLOATS / 256, 256, 0, stream>>>(W_M, Wt);
    fused_kernel<<<TOTAL_TILES / WAVES_PER_BLOCK, 32 * WAVES_PER_BLOCK, 0, stream>>>(
        feature, filters, b_M, Kt, Wt, out);
}